// LSTMModel_No_Image_17411797418021
// MI455X (gfx1250) — compile-verified
//
#include <hip/hip_runtime.h>

// MI455X (gfx1250) persistent single-workgroup LSTM model.
// 4 wave32s; fp32 WMMA (v_wmma_f32_16x16x4_f32) for the T-parallel GEMMs
// (front MLP + LSTM input-gate precompute), VALU for the serial 60-step
// recurrence. All operands zero-padded into LDS so the WMMA inner loops are
// branch-free 64-bit loads (no exec-mask divergence in the hot loops).

#define LN_EPS 1e-5f
#define LDP 66              // padded row stride (even -> 8B aligned, no bank conflicts)
#define SEG (64 * LDP)      // 4224 floats per padded buffer

typedef float v2f __attribute__((ext_vector_type(2)));
typedef float v8f __attribute__((ext_vector_type(8)));

__device__ __forceinline__ float sigm(float v) { return 1.0f / (1.0f + __expf(-v)); }
__device__ __forceinline__ float relu(float v) { return v > 0.0f ? v : 0.0f; }

// Zero-padded copy: dst is [64][LDP], src is [R][C] row-major.
// Unconditional clamped-index load + select => no divergent branches.
__device__ __forceinline__ void fill_pad(float* dst, const float* __restrict__ src,
                                         int R, int C) {
    for (int i = threadIdx.x; i < 64 * LDP; i += 128) {
        const int r = i / LDP;
        const int c = i - r * LDP;
        const int rs = (r < R) ? r : 0;
        const int cs = (c < C) ? c : 0;
        const float v = src[rs * C + cs];
        dst[i] = (r < R && c < C) ? v : 0.0f;
    }
}

__global__ __launch_bounds__(128, 1)
void lstm_model_kernel(
    const float* __restrict__ x,                                   // [60,63]
    const float* __restrict__ w00, const float* __restrict__ b00,  // [63,63],[63]
    const float* __restrict__ w01, const float* __restrict__ b01,  // [64,63],[64]
    const float* __restrict__ ln_g, const float* __restrict__ ln_b,// [64],[64]
    const float* __restrict__ w_ih, const float* __restrict__ w_hh,// [128,64],[128,32]
    const float* __restrict__ b_ih, const float* __restrict__ b_hh,// [128],[128]
    const float* __restrict__ bn_g, const float* __restrict__ bn_b,// [32],[32]
    const float* __restrict__ w10, const float* __restrict__ b10,  // [32,32],[32]
    const float* __restrict__ w11, const float* __restrict__ b11,  // [32,32],[32]
    const float* __restrict__ w12, const float* __restrict__ b12,  // [256,32],[256]
    float* __restrict__ out)                                       // [256]
{
    // 3 x 4224 floats = 50688 B pool + small state arrays.
    __shared__ __align__(16) float pool[3 * SEG];
    __shared__ float sGate[128];
    __shared__ float sH[32];
    __shared__ float sC[32];
    __shared__ float sA[32];
    __shared__ float sB[32];

    float* const buf0 = pool;            // Xpad -> F
    float* const buf1 = pool + SEG;      // W00pad -> W01pad -> (part of G)
    float* const buf2 = pool + 2 * SEG;  // Y -> (part of G)
    float* const gbuf = pool + SEG;      // G[60][128], 7680 floats (fits in buf1+buf2)

    const int tid  = threadIdx.x;
    const int wave = tid >> 5;        // 0..3 -> M-tile
    const int lane = tid & 31;
    const int half = lane >> 4;       // half-wave: element k-offset = 2*half
    const int lrow = lane & 15;
    const int m0   = wave << 4;
    const int ar   = m0 + lrow;       // A-matrix row owned by this lane
    const int k2h  = half << 1;       // fragment pair k offset

    // ---- Stage X and w00 (zero-padded) --------------------------------------
    fill_pad(buf0, x,   60, 63);
    fill_pad(buf1, w00, 63, 63);
    __syncthreads();

    // ---- Phase 1a: Y = relu(Xpad @ W00pad^T + b00)  -> buf2 -----------------
    for (int nt = 0; nt < 4; ++nt) {
        const int col = (nt << 4) + lrow;
        v8f acc = {0.f, 0.f, 0.f, 0.f, 0.f, 0.f, 0.f, 0.f};
#pragma unroll 4
        for (int k0 = 0; k0 < 64; k0 += 4) {
            const v2f a = *(const v2f*)(buf0 + ar * LDP + k0 + k2h);
            const v2f b = *(const v2f*)(buf1 + col * LDP + k0 + k2h);
            acc = __builtin_amdgcn_wmma_f32_16x16x4_f32(
                false, a, false, b, (short)0, acc, false, false);
        }
        float bias = b00[(col < 63) ? col : 0];
        bias = (col < 63) ? bias : 0.0f;
#pragma unroll
        for (int r = 0; r < 8; ++r)
            buf2[(m0 + r + (half << 3)) * LDP + col] = relu(acc[r] + bias);
    }
    __syncthreads();

    // ---- Stage w01 (zero-padded) over dead W00pad ---------------------------
    fill_pad(buf1, w01, 64, 63);
    __syncthreads();

    // ---- Phase 1b: F = relu(Y @ W01pad^T + b01)  -> buf0 --------------------
    for (int nt = 0; nt < 4; ++nt) {
        const int col = (nt << 4) + lrow;
        v8f acc = {0.f, 0.f, 0.f, 0.f, 0.f, 0.f, 0.f, 0.f};
#pragma unroll 4
        for (int k0 = 0; k0 < 64; k0 += 4) {
            const v2f a = *(const v2f*)(buf2 + ar * LDP + k0 + k2h);
            const v2f b = *(const v2f*)(buf1 + col * LDP + k0 + k2h);
            acc = __builtin_amdgcn_wmma_f32_16x16x4_f32(
                false, a, false, b, (short)0, acc, false, false);
        }
        const float bias = b01[col];
#pragma unroll
        for (int r = 0; r < 8; ++r)
            buf0[(m0 + r + (half << 3)) * LDP + col] = relu(acc[r] + bias);
    }
    __syncthreads();

    // ---- Phase 1c: per-row LayerNorm over 64 cols, in place on F ------------
    if (lane < 16) {
        float* row = buf0 + (m0 + lane) * LDP;
        float s = 0.0f;
        for (int k = 0; k < 64; ++k) s += row[k];
        const float mu = s * (1.0f / 64.0f);
        float var = 0.0f;
        for (int k = 0; k < 64; ++k) { const float d = row[k] - mu; var += d * d; }
        const float inv = rsqrtf(var * (1.0f / 64.0f) + LN_EPS);
        for (int k = 0; k < 64; ++k)
            row[k] = (row[k] - mu) * inv * ln_g[k] + ln_b[k];
    }
    __syncthreads();

    // ---- Phase 1d: G = F @ w_ih^T + b_ih + b_hh  -> gbuf[60][128] -----------
    // B fragment is a direct 8B-aligned global v2f load (K=64, N=128 exact).
    for (int nt = 0; nt < 8; ++nt) {
        const int col = (nt << 4) + lrow;
        v8f acc = {0.f, 0.f, 0.f, 0.f, 0.f, 0.f, 0.f, 0.f};
#pragma unroll 4
        for (int k0 = 0; k0 < 64; k0 += 4) {
            const v2f a = *(const v2f*)(buf0 + ar * LDP + k0 + k2h);
            const v2f b = *(const v2f*)(w_ih + col * 64 + k0 + k2h);
            acc = __builtin_amdgcn_wmma_f32_16x16x4_f32(
                false, a, false, b, (short)0, acc, false, false);
        }
        const float bias = b_ih[col] + b_hh[col];
#pragma unroll
        for (int r = 0; r < 8; ++r) {
            const int row = m0 + r + (half << 3);
            if (row < 60) gbuf[row * 128 + col] = acc[r] + bias;
        }
    }

    // Private w_hh row into registers (one-time load; hot loop stays in-reg).
    float wh[32];
#pragma unroll
    for (int k = 0; k < 32; ++k) wh[k] = w_hh[tid * 32 + k];

    if (tid < 32) { sH[tid] = 0.0f; sC[tid] = 0.0f; }
    __syncthreads();

    // ---- Phase 2: serial LSTM recurrence over T=60 --------------------------
    for (int t = 0; t < 60; ++t) {
        float g = gbuf[t * 128 + tid];
#pragma unroll
        for (int k = 0; k < 32; ++k) g += sH[k] * wh[k];
        sGate[tid] = g;
        __syncthreads();
        if (tid < 32) {
            const float ig = sGate[tid];
            const float fg = sGate[32 + tid];
            const float gg = sGate[64 + tid];
            const float og = sGate[96 + tid];
            const float c = sigm(fg) * sC[tid] + sigm(ig) * tanhf(gg);
            sC[tid] = c;
            sH[tid] = sigm(og) * tanhf(c);
        }
        __syncthreads();
    }

    // ---- Phase 3: BatchNorm(eval) + head MLP --------------------------------
    if (tid < 32)
        sA[tid] = sH[tid] * rsqrtf(1.0f + LN_EPS) * bn_g[tid] + bn_b[tid];
    __syncthreads();
    if (tid < 32) {
        float a = b10[tid];
        for (int k = 0; k < 32; ++k) a += sA[k] * w10[tid * 32 + k];
        sB[tid] = relu(a);
    }
    __syncthreads();
    if (tid < 32) {
        float a = b11[tid];
        for (int k = 0; k < 32; ++k) a += sB[k] * w11[tid * 32 + k];
        sGate[tid] = relu(a);
    }
    __syncthreads();
    for (int j = tid; j < 256; j += 128) {
        float a = b12[j];
        for (int k = 0; k < 32; ++k) a += sGate[k] * w12[j * 32 + k];
        out[j] = a;
    }
}

extern "C" void kernel_launch(void* const* d_in, const int* in_sizes, int n_in,
                              void* d_out, int out_size, void* d_ws, size_t ws_size,
                              hipStream_t stream) {
    (void)in_sizes; (void)n_in; (void)out_size; (void)d_ws; (void)ws_size;
    lstm_model_kernel<<<1, 128, 0, stream>>>(
        (const float*)d_in[0],                          // x
        (const float*)d_in[1],  (const float*)d_in[2],  // w00, b00
        (const float*)d_in[3],  (const float*)d_in[4],  // w01, b01
        (const float*)d_in[5],  (const float*)d_in[6],  // ln_g, ln_b
        (const float*)d_in[7],  (const float*)d_in[8],  // w_ih, w_hh
        (const float*)d_in[9],  (const float*)d_in[10], // b_ih, b_hh
        (const float*)d_in[11], (const float*)d_in[12], // bn_g, bn_b
        (const float*)d_in[13], (const float*)d_in[14], // w10, b10
        (const float*)d_in[15], (const float*)d_in[16], // w11, b11
        (const float*)d_in[17], (const float*)d_in[18], // w12, b12
        (float*)d_out);
}